// SharedGraphEncoder_68169720922652
// MI455X (gfx1250) — compile-verified
//
#include <hip/hip_runtime.h>

#define HIDDEN 128
#define WPAD   132  // padded LDS row stride: 132 % 64 == 4 -> conflict-free B reads

typedef __attribute__((ext_vector_type(2))) float v2f;
typedef __attribute__((ext_vector_type(8))) float v8f;

// ---------------------------------------------------------------------------
// Zero a float buffer (capture-safe replacement for hipMemsetAsync).
// ---------------------------------------------------------------------------
__global__ void zero_kernel(float* __restrict__ p, long long n) {
    long long i = (long long)blockIdx.x * blockDim.x + threadIdx.x;
    long long stride = (long long)gridDim.x * blockDim.x;
    for (; i < n; i += stride) p[i] = 0.0f;
}

// ---------------------------------------------------------------------------
// Degree count: cnt[dst[e]] += 1.0f
// ---------------------------------------------------------------------------
__global__ void count_kernel(const int* __restrict__ dst, int E,
                             float* __restrict__ cnt) {
    int i = blockIdx.x * blockDim.x + threadIdx.x;
    int stride = gridDim.x * blockDim.x;
    for (; i < E; i += stride) atomicAdd(&cnt[dst[i]], 1.0f);
}

// ---------------------------------------------------------------------------
// Edge scatter: agg[dst[e]] += feat[src[e]]  (32 lanes x float4 = 128 floats)
// Row read is a coalesced 512B burst per wave; adds land in L2 (working set
// 25.6MB << 192MB L2).
// ---------------------------------------------------------------------------
__global__ void scatter_kernel(const float* __restrict__ feat,
                               const int* __restrict__ src,
                               const int* __restrict__ dst,
                               float* __restrict__ agg, int E) {
    long long t = (long long)blockIdx.x * blockDim.x + threadIdx.x;
    int e = (int)(t >> 5);
    if (e >= E) return;
    int chunk = (int)(t & 31);
    const float4 v =
        *(const float4*)(feat + (long long)src[e] * HIDDEN + chunk * 4);
    float* outp = agg + (long long)dst[e] * HIDDEN + chunk * 4;
    atomicAdd(outp + 0, v.x);
    atomicAdd(outp + 1, v.y);
    atomicAdd(outp + 2, v.z);
    atomicAdd(outp + 3, v.w);
}

// ---------------------------------------------------------------------------
// Fused SAGE GEMM using V_WMMA_F32_16X16X4_F32:
//   out[m][n] = (agg[m]/max(cnt[m],1)) . Wl[n] + self[m] . Wr[n] + bias[n]
// One wave per 16-row M-tile; 8 v8f accumulators cover N=128.
//
// Both weight matrices are staged into LDS once per block (2 x 128 x 132
// floats = 132KB; 2 blocks/WGP within the 320KB CDNA5 LDS). Row padding of
// 4 floats makes the 16-lane B-fragment ds_load_b64 bank-conflict-free.
//
// A 16x4 f32 fragment layout (ISA 7.12.2): lanes 0-15 hold (K=k0,k0+1) for
// M=lane; lanes 16-31 hold (K=k0+2,k0+3) for M=lane-16.
// B 4x16 fragment mirrors it across N. C/D: VGPR v -> M = v + 8*(lane>=16),
// N = lane&15 + 16*nt.
// ---------------------------------------------------------------------------
__global__ void __launch_bounds__(256) sage_gemm_kernel(
    const float* __restrict__ self_feat, const float* __restrict__ agg,
    const float* __restrict__ cnt, const float* __restrict__ Wl,
    const float* __restrict__ Wr, const float* __restrict__ bias,
    float* __restrict__ out, int n_nodes, int do_relu) {
    __shared__ float sW[2][HIDDEN][WPAD];

    // Cooperative stage-in of Wl and Wr (float4 granularity, 16B aligned:
    // row stride 132*4 = 528 bytes, multiple of 16).
    for (int idx = threadIdx.x; idx < (HIDDEN * HIDDEN) / 4; idx += 256) {
        int row  = idx >> 5;        // 32 float4 chunks per 128-float row
        int col4 = (idx & 31) << 2;
        float4 vl = *(const float4*)(Wl + (long long)row * HIDDEN + col4);
        float4 vr = *(const float4*)(Wr + (long long)row * HIDDEN + col4);
        *(float4*)(&sW[0][row][col4]) = vl;
        *(float4*)(&sW[1][row][col4]) = vr;
    }
    __syncthreads();

    const int lane  = threadIdx.x & 31;
    const int tileM = blockIdx.x * (blockDim.x >> 5) + (threadIdx.x >> 5);
    if (tileM * 16 < n_nodes) {  // wave-uniform guard: EXEC all-1 inside WMMA
        const int half = lane >> 4;  // 0 -> K pair {k0,k0+1}, 1 -> {k0+2,k0+3}
        const int r    = lane & 15;  // A-row / B-col within tile
        const int m    = tileM * 16 + r;
        const int kOff = half * 2;

        const float invDeg = 1.0f / fmaxf(cnt[m], 1.0f);

        v8f acc[8] = {};

        const float* aggRow  = agg + (long long)m * HIDDEN + kOff;
        const float* selfRow = self_feat + (long long)m * HIDDEN + kOff;

#pragma unroll 1
        for (int k0 = 0; k0 < HIDDEN; k0 += 4) {
            v2f aA = *(const v2f*)(aggRow + k0);
            aA.x *= invDeg;
            aA.y *= invDeg;
            v2f aX = *(const v2f*)(selfRow + k0);
#pragma unroll
            for (int nt = 0; nt < 8; ++nt) {
                const int n = nt * 16 + r;
                v2f bL = *(const v2f*)(&sW[0][n][k0 + kOff]);
                v2f bR = *(const v2f*)(&sW[1][n][k0 + kOff]);
                acc[nt] = __builtin_amdgcn_wmma_f32_16x16x4_f32(
                    false, aA, false, bL, (short)0, acc[nt], false, false);
                acc[nt] = __builtin_amdgcn_wmma_f32_16x16x4_f32(
                    false, aX, false, bR, (short)0, acc[nt], false, false);
            }
        }

#pragma unroll
        for (int nt = 0; nt < 8; ++nt) {
            const int n = nt * 16 + r;
            const float b = bias[n];
#pragma unroll
            for (int v = 0; v < 8; ++v) {
                int row = tileM * 16 + v + 8 * half;
                float val = acc[nt][v] + b;
                if (do_relu) val = fmaxf(val, 0.0f);
                out[(long long)row * HIDDEN + n] = val;
            }
        }
    }
}

// ---------------------------------------------------------------------------
// Pipeline:
//   zero(cnt,agg) -> count -> scatter(x) -> gemm1(relu) -> h
//   zero(agg)     ->          scatter(h) -> gemm2       -> out
// ---------------------------------------------------------------------------
extern "C" void kernel_launch(void* const* d_in, const int* in_sizes, int n_in,
                              void* d_out, int out_size, void* d_ws,
                              size_t ws_size, hipStream_t stream) {
    const float* x   = (const float*)d_in[0];
    const int*   ei  = (const int*)d_in[1];
    const float* W1l = (const float*)d_in[2];
    const float* b1l = (const float*)d_in[3];
    const float* W1r = (const float*)d_in[4];
    const float* W2l = (const float*)d_in[5];
    const float* b2l = (const float*)d_in[6];
    const float* W2r = (const float*)d_in[7];
    float* out = (float*)d_out;

    const int N = in_sizes[0] / HIDDEN;  // 50000
    const int E = in_sizes[1] / 2;       // 800000
    const int* src = ei;
    const int* dst = ei + E;

    // Workspace layout (floats): cnt[N] | agg[N*128] | h[N*128]  (~51.4 MB)
    float* ws  = (float*)d_ws;
    float* cnt = ws;
    float* agg = ws + N;
    float* h   = agg + (long long)N * HIDDEN;

    const long long aggN = (long long)N * HIDDEN;

    // Layer 1
    zero_kernel<<<4096, 256, 0, stream>>>(cnt, (long long)N + aggN);
    count_kernel<<<(E + 255) / 256, 256, 0, stream>>>(dst, E, cnt);
    {
        long long st = (long long)E * 32;
        int sblocks = (int)((st + 255) / 256);
        scatter_kernel<<<sblocks, 256, 0, stream>>>(x, src, dst, agg, E);
    }
    const int mtiles  = (N + 15) / 16;
    const int gblocks = (mtiles + 7) / 8;
    sage_gemm_kernel<<<gblocks, 256, 0, stream>>>(x, agg, cnt, W1l, W1r, b1l,
                                                  h, N, 1);

    // Layer 2
    zero_kernel<<<4096, 256, 0, stream>>>(agg, aggN);
    {
        long long st = (long long)E * 32;
        int sblocks = (int)((st + 255) / 256);
        scatter_kernel<<<sblocks, 256, 0, stream>>>(h, src, dst, agg, E);
    }
    sage_gemm_kernel<<<gblocks, 256, 0, stream>>>(h, agg, cnt, W2l, W2r, b2l,
                                                  out, N, 0);
}